// GraphConv_67250597921520
// MI455X (gfx1250) — compile-verified
//
#include <hip/hip_runtime.h>
#include <float.h>

typedef __attribute__((ext_vector_type(2))) float v2f;
typedef __attribute__((ext_vector_type(8))) float v8f;

#define GN 8192
#define GF 128

// C[16x16 per wave] = A[M x K] * B[K x 128] (+bias, +relu) using V_WMMA_F32_16X16X4_F32.
// Block = 256 threads = 8 waves; wave w owns column tile w (16 cols), block owns 16 rows.
__global__ __launch_bounds__(256) void wmma_gemm_f32(
    const float* __restrict__ A, const float* __restrict__ B,
    const float* __restrict__ bias, float* __restrict__ C,
    int K, int doRelu)
{
    const int lane = threadIdx.x & 31;
    const int wave = threadIdx.x >> 5;
    const int half = lane >> 4;   // 0: lanes 0-15, 1: lanes 16-31
    const int l    = lane & 15;
    const int rowBase = blockIdx.x * 16;
    const int colBase = wave * 16;

    // A fragment: lane l%16 -> row M; VGPR0/1 hold K = 2*half + {0,1} -> contiguous b64 load
    const float* aRow  = A + (size_t)(rowBase + l) * (size_t)K + 2 * half;
    // B fragment: lane l%16 -> col N; VGPR0/1 hold K rows (2*half, 2*half+1)
    const float* bBase = B + (size_t)(2 * half) * GF + colBase + l;

    v8f acc = {};

#pragma unroll 8
    for (int k = 0; k < K; k += 4) {
        v2f a = *(const v2f*)(aRow + (size_t)k);
        const float* bp = bBase + (size_t)k * GF;
        v2f b;
        b.x = bp[0];
        b.y = bp[GF];
        // (neg_a, A, neg_b, B, c_mod, C, reuse_a, reuse_b)
        acc = __builtin_amdgcn_wmma_f32_16x16x4_f32(
            false, a, false, b, (short)0, acc, false, false);
    }

    const float bv = bias ? bias[colBase + l] : 0.0f;
    float* cp = C + (size_t)(rowBase + half * 8) * GF + colBase + l;
#pragma unroll
    for (int v = 0; v < 8; ++v) {
        float r = acc[v] + bv;                 // C layout: VGPR v -> row v + 8*half
        if (doRelu) r = fmaxf(r, 0.0f);
        cp[(size_t)v * GF] = r;
    }
}

// Per-row masked neighbor max-pool: res[i,f] = max_{j: adj[i,j]!=0} out[j,f].
// One wave per row (wave32): lanes coalesce-scan 32 adj entries, ballot the
// nonzeros, then gather out[j] rows (4 floats/lane = b128, 512B coalesced).
__global__ __launch_bounds__(256) void graph_pool(
    const float* __restrict__ adj, const float* __restrict__ out,
    float* __restrict__ res)
{
    const int lane = threadIdx.x & 31;
    const int wave = threadIdx.x >> 5;
    const int row  = blockIdx.x * 8 + wave;

    const float* arow = adj + (size_t)row * GN;
    float4 m = make_float4(-FLT_MAX, -FLT_MAX, -FLT_MAX, -FLT_MAX);

    for (int j0 = 0; j0 < GN; j0 += 32) {
        float av = arow[j0 + lane];
        unsigned long long bm = __ballot(av != 0.0f);   // wave32: low 32 bits
        while (bm) {
            int b = __builtin_ctzll(bm);
            bm &= bm - 1;
            const float4 v = *(const float4*)(out + (size_t)(j0 + b) * GF + lane * 4);
            m.x = fmaxf(m.x, v.x);
            m.y = fmaxf(m.y, v.y);
            m.z = fmaxf(m.z, v.z);
            m.w = fmaxf(m.w, v.w);
        }
    }
    *(float4*)(res + (size_t)row * GF + lane * 4) = m;
}

extern "C" void kernel_launch(void* const* d_in, const int* in_sizes, int n_in,
                              void* d_out, int out_size, void* d_ws, size_t ws_size,
                              hipStream_t stream) {
    const float* x    = (const float*)d_in[0];   // [8192,128]
    const float* adj  = (const float*)d_in[1];   // [8192,8192]
    const float* w    = (const float*)d_in[2];   // [128,128]
    const float* bias = (const float*)d_in[3];   // [128]
    float* res = (float*)d_out;                  // [8192,128]

    float* h   = (float*)d_ws;                   // [8192,128]  (4 MB)
    float* out = h + (size_t)GN * GF;            // [8192,128]  (4 MB)

    // h = x @ w
    wmma_gemm_f32<<<GN / 16, 256, 0, stream>>>(x, w, nullptr, h, GF, 0);
    // out = relu(adj @ h + bias)   (memory-bound: streams 256 MB of adj)
    wmma_gemm_f32<<<GN / 16, 256, 0, stream>>>(adj, h, bias, out, GN, 1);
    // res = neighbor max-pool      (adj stream of 64 MB: 2048 rows; out gathers hit L2)
    graph_pool<<<GN / 32, 256, 0, stream>>>(adj, out, res);
}